// LFLF_GCN_39814346834049
// MI455X (gfx1250) — compile-verified
//
#include <hip/hip_runtime.h>
#include <hip/hip_bf16.h>
#include <stdint.h>

typedef __attribute__((ext_vector_type(2))) float v2f;
typedef __attribute__((ext_vector_type(8))) float v8f;

#define NN 50000
#define EE 800000
#define HID 128
#define CLS 64
#define ATT_H 16

#if defined(__gfx1250__)
#  define USE_ASYNC_LDS 1
#else
#  define USE_ASYNC_LDS 0
#endif

// ---------------------------------------------------------------------------
// 16B global -> LDS copy. On gfx1250 use the async Tensor/LDS DMA path
// (GLOBAL_LOAD_ASYNC_TO_LDS_B128, tracked by ASYNCcnt): no VGPR round-trip.
// Generic pointers to LDS carry the LDS byte address in their low 32 bits
// (ISA 10.2 aperture rules), so truncation gives the VDST LDS address.
// ---------------------------------------------------------------------------
__device__ __forceinline__ void cp_lds_f4(float* dst, const float* src) {
#if USE_ASYNC_LDS
    asm volatile("global_load_async_to_lds_b128 %0, %1, off"
                 :
                 : "v"((uint32_t)(uintptr_t)dst),
                   "v"((uint64_t)(uintptr_t)src)
                 : "memory");
#else
    *(float4*)dst = *(const float4*)src;
#endif
}

__device__ __forceinline__ void cp_lds_wait_barrier() {
#if USE_ASYNC_LDS
    asm volatile("s_wait_asynccnt 0x0" ::: "memory");
#endif
    __syncthreads();
}

// ---------------------------------------------------------------------------
// init: degA/degB = 1 (self loop), accA/accB = 0
// ---------------------------------------------------------------------------
__global__ void init_ws_kernel(float* __restrict__ accA, float* __restrict__ accB,
                               float* __restrict__ degA, float* __restrict__ degB, int n) {
    int i = blockIdx.x * blockDim.x + threadIdx.x;
    int total = n * HID;
    if (i < total) { accA[i] = 0.0f; accB[i] = 0.0f; }
    if (i < n)     { degA[i] = 1.0f; degB[i] = 1.0f; }
}

// ---------------------------------------------------------------------------
// edge degrees: degA[c] += 1 (ones conv), degB[c] += ew (weighted conv)
// ---------------------------------------------------------------------------
__global__ void edge_deg_kernel(const int* __restrict__ col, const float* __restrict__ ew,
                                float* __restrict__ degA, float* __restrict__ degB, int e) {
    int i = blockIdx.x * blockDim.x + threadIdx.x;
    if (i < e) {
        int c = col[i];
        atomicAdd(&degA[c], 1.0f);
        atomicAdd(&degB[c], ew[i]);
    }
}

// ---------------------------------------------------------------------------
// WMMA f32 GEMM: C[rows x M] = act(A[rows x K] @ W[K x M] + bias)
// grid.x = rows/16 (rows multiple of 16); block = 32*(M/16) threads.
// Wave w computes the 16x16 tile at cols [16w, 16w+16).
// A tile and a 64-row k-slab of W are staged in LDS via async-to-LDS copies.
// act: 0 = none, 1 = sigmoid
// ---------------------------------------------------------------------------
template <int K, int M>
__global__ void gemm_wmma_kernel(const float* __restrict__ A, const float* __restrict__ W,
                                 const float* __restrict__ bias, float* __restrict__ C,
                                 int act) {
    constexpr int KP = K + 4;                 // padded LDS stride for A
    constexpr int KB = (K > 64) ? 64 : K;     // W k-slab staged per barrier
    __shared__ float As[16 * KP];
    __shared__ float Ws[KB * M];

    const int tile = blockIdx.x;

    // stage the 16 x K A-tile (async b128 per lane)
    for (int idx = threadIdx.x; idx < 16 * (K / 4); idx += blockDim.x) {
        const int r  = idx / (K / 4);
        const int c4 = (idx % (K / 4)) * 4;
        cp_lds_f4(&As[r * KP + c4], A + ((size_t)tile * 16 + r) * K + c4);
    }

    const int wave = threadIdx.x >> 5;
    const int lane = threadIdx.x & 31;
    const int col  = wave * 16 + (lane & 15);
    const int koff = (lane < 16) ? 0 : 2;     // ISA f32 A/B fragment half-wave K offset
    const float* As_row = &As[(lane & 15) * KP];

    v8f acc = {};
    for (int kb = 0; kb < K; kb += KB) {
        // stage KB x M slab of W
        for (int idx = threadIdx.x; idx < (KB * M) / 4; idx += blockDim.x) {
            cp_lds_f4(&Ws[idx * 4], W + (size_t)kb * M + idx * 4);
        }
        cp_lds_wait_barrier();

        for (int k0 = 0; k0 < KB; k0 += 4) {
            const int kk = k0 + koff;
            v2f a; a.x = As_row[kb + kk];        a.y = As_row[kb + kk + 1];
            v2f b; b.x = Ws[kk * M + col];       b.y = Ws[(kk + 1) * M + col];
            acc = __builtin_amdgcn_wmma_f32_16x16x4_f32(false, a, false, b, (short)0, acc,
                                                        false, false);
        }
        __syncthreads();   // protect Ws before the next slab overwrites it
    }

    // C layout: lanes 0-15 rows j, lanes 16-31 rows 8+j (per VGPR j)
    const int rbase = tile * 16 + ((lane >> 4) ? 8 : 0);
    const float bv = bias ? bias[col] : 0.0f;
#pragma unroll
    for (int j = 0; j < 8; ++j) {
        float v = acc[j] + bv;
        if (act == 1) v = 1.0f / (1.0f + __expf(-v));
        C[(size_t)(rbase + j) * M + col] = v;
    }
}

// ---------------------------------------------------------------------------
// scatter: wave per edge; both convs share row/col so a single wave serves both.
// accA[c] += dA(r)*dA(c) * xw[r];  accB[c] += dB(r)*w*dB(c) * yw[r]
// ---------------------------------------------------------------------------
__global__ void scatter_kernel(const int* __restrict__ row, const int* __restrict__ col,
                               const float* __restrict__ ew,
                               const float* __restrict__ xw, const float* __restrict__ yw,
                               const float* __restrict__ degA, const float* __restrict__ degB,
                               float* __restrict__ accA, float* __restrict__ accB, int e) {
    const int wv   = (blockIdx.x * blockDim.x + threadIdx.x) >> 5;
    const int lane = threadIdx.x & 31;
    if (wv >= e) return;
    const int r = row[wv];
    const int c = col[wv];
    const float w  = ew[wv];
    const float na = __frsqrt_rn(degA[r]) * __frsqrt_rn(degA[c]);
    const float nb = __frsqrt_rn(degB[r]) * w * __frsqrt_rn(degB[c]);

    const size_t rb = (size_t)r * HID + lane * 4;
    const size_t cb = (size_t)c * HID + lane * 4;
    const float4 xa = *(const float4*)(xw + rb);
    const float4 ya = *(const float4*)(yw + rb);
    float* pa = accA + cb;
    float* pb = accB + cb;
    atomicAdd(pa + 0, na * xa.x); atomicAdd(pa + 1, na * xa.y);
    atomicAdd(pa + 2, na * xa.z); atomicAdd(pa + 3, na * xa.w);
    atomicAdd(pb + 0, nb * ya.x); atomicAdd(pb + 1, nb * ya.y);
    atomicAdd(pb + 2, nb * ya.z); atomicAdd(pb + 3, nb * ya.w);
}

// ---------------------------------------------------------------------------
// fused epilogue: wave per node (wave32). Adds self-loop term + bias, computes
// AM-GCN attention over the two views, softmax fuse, optional relu, writes x.
// ---------------------------------------------------------------------------
__global__ void fuse_kernel(const float* __restrict__ accA, const float* __restrict__ accB,
                            const float* __restrict__ xw, const float* __restrict__ yw,
                            const float* __restrict__ degA, const float* __restrict__ degB,
                            const float* __restrict__ bA, const float* __restrict__ bB,
                            const float* __restrict__ ap1, const float* __restrict__ apb1,
                            const float* __restrict__ ap2,
                            float* __restrict__ xout, int n, int do_relu) {
    __shared__ float sAp1[HID * ATT_H];
    __shared__ float sApb1[ATT_H];
    __shared__ float sAp2[ATT_H];
    for (int i = threadIdx.x; i < (HID * ATT_H) / 4; i += blockDim.x)
        cp_lds_f4(&sAp1[i * 4], ap1 + i * 4);
    if (threadIdx.x < ATT_H) {
        sApb1[threadIdx.x] = apb1[threadIdx.x];
        sAp2[threadIdx.x]  = ap2[threadIdx.x];
    }
    cp_lds_wait_barrier();

    const int wave = threadIdx.x >> 5;
    const int lane = threadIdx.x & 31;
    const int node = blockIdx.x * 8 + wave;   // blockDim.x == 256 -> 8 waves
    if (node >= n) return;

    const size_t base = (size_t)node * HID + lane * 4;
    const float invA = 1.0f / degA[node];     // dinv^2 for self-loop weight 1
    const float invB = 1.0f / degB[node];

    float ea[4], eb[4];
    {
        const float4 aA = *(const float4*)(accA + base);
        const float4 xA = *(const float4*)(xw + base);
        const float4 aB = *(const float4*)(accB + base);
        const float4 yA = *(const float4*)(yw + base);
        const int kb = lane * 4;
        ea[0] = aA.x + invA * xA.x + bA[kb + 0];
        ea[1] = aA.y + invA * xA.y + bA[kb + 1];
        ea[2] = aA.z + invA * xA.z + bA[kb + 2];
        ea[3] = aA.w + invA * xA.w + bA[kb + 3];
        eb[0] = aB.x + invB * yA.x + bB[kb + 0];
        eb[1] = aB.y + invB * yA.y + bB[kb + 1];
        eb[2] = aB.z + invB * yA.z + bB[kb + 2];
        eb[3] = aB.w + invB * yA.w + bB[kb + 3];
    }

    // attention scores: s_j = sum_k emb[k] * ap1[k][j]
    float sa[ATT_H], sb[ATT_H];
#pragma unroll
    for (int j = 0; j < ATT_H; ++j) { sa[j] = 0.0f; sb[j] = 0.0f; }
#pragma unroll
    for (int i = 0; i < 4; ++i) {
        const int k = lane * 4 + i;
#pragma unroll
        for (int j = 0; j < ATT_H; ++j) {
            const float p = sAp1[k * ATT_H + j];
            sa[j] += ea[i] * p;
            sb[j] += eb[i] * p;
        }
    }
    // wave32 butterfly reduction -> all lanes hold the full sums
#pragma unroll
    for (int m = 16; m >= 1; m >>= 1) {
#pragma unroll
        for (int j = 0; j < ATT_H; ++j) {
            sa[j] += __shfl_xor(sa[j], m, 32);
            sb[j] += __shfl_xor(sb[j], m, 32);
        }
    }

    float wA = 0.0f, wB = 0.0f;
#pragma unroll
    for (int j = 0; j < ATT_H; ++j) {
        wA += tanhf(sa[j] + sApb1[j]) * sAp2[j];
        wB += tanhf(sb[j] + sApb1[j]) * sAp2[j];
    }
    const float mx = fmaxf(wA, wB);
    const float eAx = __expf(wA - mx), eBx = __expf(wB - mx);
    const float inv = 1.0f / (eAx + eBx);
    const float betaA = eAx * inv, betaB = eBx * inv;

    float4 o;
    o.x = betaA * ea[0] + betaB * eb[0];
    o.y = betaA * ea[1] + betaB * eb[1];
    o.z = betaA * ea[2] + betaB * eb[2];
    o.w = betaA * ea[3] + betaB * eb[3];
    if (do_relu) {
        o.x = fmaxf(o.x, 0.0f); o.y = fmaxf(o.y, 0.0f);
        o.z = fmaxf(o.z, 0.0f); o.w = fmaxf(o.w, 0.0f);
    }
    *(float4*)(xout + base) = o;
}

// ---------------------------------------------------------------------------
extern "C" void kernel_launch(void* const* d_in, const int* in_sizes, int n_in,
                              void* d_out, int out_size, void* d_ws, size_t ws_size,
                              hipStream_t stream) {
    const float* x    = (const float*)d_in[0];   // [N,128]
    const float* y    = (const float*)d_in[1];   // [N,64]
    const int*   row  = (const int*)d_in[2];     // [E]
    const int*   col  = (const int*)d_in[3];     // [E]
    const float* ew   = (const float*)d_in[4];   // [2,E]
    const float* W0   = (const float*)d_in[5];
    const float* b0   = (const float*)d_in[6];
    const float* W1   = (const float*)d_in[7];
    const float* b1   = (const float*)d_in[8];
    const float* Wl   = (const float*)d_in[9];
    const float* bl   = (const float*)d_in[10];
    const float* ap1  = (const float*)d_in[11];
    const float* apb1 = (const float*)d_in[12];
    const float* ap2  = (const float*)d_in[13];
    const float* mW   = (const float*)d_in[14];
    const float* mb   = (const float*)d_in[15];

    float* xout = (float*)d_out;                    // [N,128]
    float* yout = xout + (size_t)NN * HID;          // [N,64]

    float* ws   = (float*)d_ws;                     // ~103 MB used
    float* xw   = ws;                               // N*128
    float* yw   = xw + (size_t)NN * HID;            // N*128
    float* accA = yw + (size_t)NN * HID;            // N*128
    float* accB = accA + (size_t)NN * HID;          // N*128
    float* degA = accB + (size_t)NN * HID;          // N
    float* degB = degA + NN;                        // N

    const int rowTiles = NN / 16;                   // 3125, exact

    for (int layer = 0; layer < 2; ++layer) {
        const float* Xin = (layer == 0) ? x : xout;
        const float* Yin = (layer == 0) ? y : yout;
        const float* Wi  = (layer == 0) ? W0 : W1;
        const float* bi  = (layer == 0) ? b0 : b1;
        const float* ewi = ew + (size_t)layer * EE;

        init_ws_kernel<<<(NN * HID + 255) / 256, 256, 0, stream>>>(accA, accB, degA, degB, NN);
        edge_deg_kernel<<<(EE + 255) / 256, 256, 0, stream>>>(col, ewi, degA, degB, EE);

        gemm_wmma_kernel<HID, HID><<<rowTiles, 256, 0, stream>>>(Xin, Wi, nullptr, xw, 0);
        gemm_wmma_kernel<CLS, HID><<<rowTiles, 256, 0, stream>>>(Yin, Wl, nullptr, yw, 0);

        scatter_kernel<<<(EE + 7) / 8, 256, 0, stream>>>(row, col, ewi, xw, yw,
                                                         degA, degB, accA, accB, EE);

        fuse_kernel<<<(NN + 7) / 8, 256, 0, stream>>>(accA, accB, xw, yw, degA, degB,
                                                      bi, bl, ap1, apb1, ap2,
                                                      xout, NN, (layer == 0) ? 1 : 0);

        gemm_wmma_kernel<HID, CLS><<<rowTiles, 128, 0, stream>>>(xout, mW, mb, yout, 1);
    }
}